// Graph_47047071761000
// MI455X (gfx1250) — compile-verified
//
#include <hip/hip_runtime.h>
#include <math.h>

#define N_FRAMES 7
#define PPF      128
#define C_CH     128
#define H_IN     256
#define W_IN     256
#define DS_F     4
#define H2       (H_IN / DS_F)   // 64
#define W2       (W_IN / DS_F)   // 64
#define SEARCH   7               // NB + PATCH - 1
#define NQ       49              // SEARCH*SEARCH
#define NQ_PAD   64              // padded N for WMMA tiles

static constexpr float PI_F   = 3.14159265358979323846f;
static constexpr float FOVH_F = 130.0f * PI_F / 180.0f;
static constexpr float FOVV_F = 20.0f  * PI_F / 180.0f;
static constexpr float RMIN_F = 0.5f;
static constexpr float RMAX_F = 30.0f;

typedef __attribute__((ext_vector_type(2))) float v2f;
typedef __attribute__((ext_vector_type(4))) float v4f;
typedef __attribute__((ext_vector_type(8))) float v8f;

// LDS partition (floats)
#define OFF_APK   0                          // 64cp x 16p x 2      = 2048
#define OFF_BPK1  (OFF_APK  + 2048)          // 64cp x 64q x 2      = 8192
#define OFF_BPK2  (OFF_BPK1 + 8192)          // 8192
#define OFF_WIN1  (OFF_BPK2 + 8192)          // 128c x 8 x 8        = 8192
#define OFF_WIN2  (OFF_WIN1 + 8192)          // 8192
#define OFF_M     (OFF_WIN2 + 8192)          // 2 x 16 x 64         = 2048
#define OFF_SH    (OFF_M    + 2048)          // 16
#define SMEM_FLOATS (OFF_SH + 16)

// ---------------- avg-pool fmap (H,W) -> fmap2 (H/4, W/4), vectorized ----------------
__global__ __launch_bounds__(256) void pool_kernel(const float* __restrict__ fmap,
                                                   float* __restrict__ fmap2, int total) {
    int idx = blockIdx.x * 256 + threadIdx.x;
    if (idx >= total) return;
    int x2 = idx % W2;
    int t  = idx / W2;
    int y2 = t % H2;
    t /= H2;                      // t = frame*C + channel
    const float* base = fmap + (size_t)t * H_IN * W_IN + (size_t)(y2 * 4) * W_IN;
    float s = 0.f;
#pragma unroll
    for (int a = 0; a < 4; ++a) {
        v4f r = *(const v4f*)(base + (size_t)a * W_IN + x2 * 4);   // 16B-aligned b128
        s += r.x + r.y + r.z + r.w;
    }
    fmap2[idx] = s * (1.0f / 16.0f);
}

// ---------------- helpers ----------------
__device__ __forceinline__ void quat_rot(const float q[4], const float v[3], float out[3]) {
    float tx = 2.f * (q[1] * v[2] - q[2] * v[1]);
    float ty = 2.f * (q[2] * v[0] - q[0] * v[2]);
    float tz = 2.f * (q[0] * v[1] - q[1] * v[0]);
    out[0] = v[0] + q[3] * tx + (q[1] * tz - q[2] * ty);
    out[1] = v[1] + q[3] * ty + (q[2] * tx - q[0] * tz);
    out[2] = v[2] + q[3] * tz + (q[0] * ty - q[1] * tx);
}

__device__ __forceinline__ float lerp2d(const float* w8x8, int i, int k, float wy, float wx) {
    float a = w8x8[i * 8 + k];
    float b = w8x8[i * 8 + k + 1];
    float c = w8x8[(i + 1) * 8 + k];
    float d = w8x8[(i + 1) * 8 + k + 1];
    return (1.f - wy) * ((1.f - wx) * a + wx * b) + wy * ((1.f - wx) * c + wx * d);
}

// ---------------- fused project + window-sample + WMMA-correlate, 1 WG/edge ----------------
__global__ __launch_bounds__(256) void corr_kernel(
    const float* __restrict__ poses,        // (7,7)
    const float* __restrict__ patch_state,  // (896,3)
    const float* __restrict__ fmap,         // (7,128,256,256)
    const float* __restrict__ patches_f,    // (896,128,4,4)
    const float* __restrict__ patches_c,    // (896,128,4,4)
    const int*   __restrict__ edge_i,
    const int*   __restrict__ edge_j,
    const float* __restrict__ fmap2,        // (7,128,64,64)  [workspace]
    float* __restrict__ out, int E) {
    extern __shared__ float smem[];
    float* Apk  = smem + OFF_APK;    // packed A: [(c>>1)*16+p]*2 + (c&1)
    float* Bpk1 = smem + OFF_BPK1;   // packed B map1: [(c>>1)*64+q]*2 + (c&1)
    float* Bpk2 = smem + OFF_BPK2;
    float* Win1 = smem + OFF_WIN1;   // 8x8 window per channel
    float* Win2 = smem + OFF_WIN2;
    float* Mm   = smem + OFF_M;      // M[map][p][q]
    float* sh   = smem + OFF_SH;     // py, px, valid

    const int e   = blockIdx.x;
    const int tid = threadIdx.x;
    const int ei  = edge_i[e];
    const int ej  = edge_j[e];

    if (tid == 0) {
        // ---- projection (replicates reference _project exactly) ----
        const float* ps = poses + (size_t)(ei / PPF) * 7;
        const float* pt = poses + (size_t)ej * 7;
        const float* st = patch_state + (size_t)ei * 3;
        float r  = st[1] / (float)H_IN * (RMAX_F - RMIN_F) + RMIN_F;
        float th = (st[0] / (float)W_IN - 0.5f) * FOVH_F * (PI_F / 180.0f);
        float ph = st[2];
        float cp = cosf(ph);
        float p3[3] = {r * cp * cosf(th), r * cp * sinf(th), r * sinf(ph)};
        float qs[4] = {ps[3], ps[4], ps[5], ps[6]};
        float qt[4] = {pt[3], pt[4], pt[5], pt[6]};
        float ns = rsqrtf(qs[0]*qs[0] + qs[1]*qs[1] + qs[2]*qs[2] + qs[3]*qs[3]);
        float nt = rsqrtf(qt[0]*qt[0] + qt[1]*qt[1] + qt[2]*qt[2] + qt[3]*qt[3]);
#pragma unroll
        for (int k = 0; k < 4; ++k) { qs[k] *= ns; qt[k] *= nt; }
        float pw[3];
        quat_rot(qs, p3, pw);
        pw[0] += ps[0]; pw[1] += ps[1]; pw[2] += ps[2];
        float qtc[4] = {-qt[0], -qt[1], -qt[2], qt[3]};
        float d3[3]  = {pw[0] - pt[0], pw[1] - pt[1], pw[2] - pt[2]};
        float pv[3];
        quat_rot(qtc, d3, pv);
        float r2   = sqrtf(pv[0]*pv[0] + pv[1]*pv[1] + pv[2]*pv[2]);
        float th2  = atan2f(pv[1], pv[0]);
        float sarg = pv[2] / fmaxf(r2, 1e-8f);
        float phi2 = asinf(fminf(fmaxf(sarg, -1.0f), 1.0f));
        bool  vld  = (r2 >= RMIN_F) && (r2 <= RMAX_F) &&
                     (fabsf(th2) <= FOVH_F * 0.5f) && (fabsf(phi2) <= FOVV_F * 0.5f);
        sh[0] = (r2 - RMIN_F) / (RMAX_F - RMIN_F) * (float)H_IN;        // py
        sh[1] = (th2 * 180.0f / PI_F / FOVH_F + 0.5f) * (float)W_IN;    // px
        sh[2] = vld ? 1.0f : 0.0f;
    }

    // ---- stage A operand into packed-pair layout (independent of projection) ----
    {
        const float* afg = patches_f + (size_t)ei * (C_CH * 16);
        for (int m = tid; m < C_CH * 16; m += 256) {
            int c = m >> 4, p = m & 15;
            Apk[(((c >> 1) << 4) + p) * 2 + (c & 1)] = afg[m];
        }
    }
    __syncthreads();

    // NaN/range-safe pixel coords (valid edges unaffected; masked edges yield zeros)
    const float pyc = fminf(fmaxf(sh[0], -100000.f), 100000.f);
    const float pxc = fminf(fmaxf(sh[1], -100000.f), 100000.f);
    const float vflg = sh[2];

    const float* img1 = fmap  + (size_t)ej * C_CH * H_IN * W_IN;
    const float* img2 = fmap2 + (size_t)ej * C_CH * H2 * W2;
    const float  s2   = (float)(W2 - 1) / (float)(W_IN - 1);   // 63/255

    // window bases (integer offsets for t1; t2 taps also fit an 8x8 window)
    const int xb1 = (int)floorf(pxc) - 3;
    const int yb1 = (int)floorf(pyc) - 3;
    const int xb2 = (int)floorf((pxc - 12.f) * s2);
    const int yb2 = (int)floorf((pyc - 12.f) * s2);

    // ---- stage 8x8 windows per channel, zero-padded at image borders ----
    for (int m = tid; m < C_CH * 64; m += 256) {
        int c = m >> 6, el = m & 63;
        int row = el >> 3, col = el & 7;
        int y1 = yb1 + row, x1 = xb1 + col;
        int y2 = yb2 + row, x2 = xb2 + col;
        float w1 = 0.f, w2 = 0.f;
        if ((unsigned)y1 < H_IN && (unsigned)x1 < W_IN)
            w1 = img1[(size_t)c * H_IN * W_IN + (size_t)y1 * W_IN + x1];
        if ((unsigned)y2 < H2 && (unsigned)x2 < W2)
            w2 = img2[(size_t)c * H2 * W2 + (size_t)y2 * W2 + x2];
        Win1[m] = w1;
        Win2[m] = w2;
    }
    __syncthreads();

    // ---- bilinear combine windows -> packed B operands ----
    const float wx1 = pxc - floorf(pxc);
    const float wy1 = pyc - floorf(pyc);
    for (int m = tid; m < C_CH * NQ_PAD; m += 256) {
        int   c = m >> 6, q = m & 63;
        float v1 = 0.f, v2 = 0.f;
        if (q < NQ) {
            int i = q / SEARCH, k = q % SEARCH;
            // map1: integer-offset grid, shared fractional weights
            v1 = lerp2d(Win1 + c * 64, i, k, wy1, wx1);
            // map2: per-position fractions inside the 8x8 window
            float xf = (pxc + 4.f * (float)(k - 3)) * s2;
            float yf = (pyc + 4.f * (float)(i - 3)) * s2;
            float x0 = floorf(xf), y0 = floorf(yf);
            int   ck = (int)x0 - xb2;   // in [0,6]
            int   ci = (int)y0 - yb2;   // in [0,6]
            v2 = lerp2d(Win2 + c * 64, ci, ck, yf - y0, xf - x0);
        }
        int pk = (((c >> 1) << 6) + q) * 2 + (c & 1);
        Bpk1[pk] = v1;
        Bpk2[pk] = v2;
    }
    __syncthreads();

    // ---- GEMM: M[p][q] = sum_c Af[c][p]*T[c][q]; 8 waves = 2 maps x 4 N-tiles ----
    {
        const int wave = tid >> 5;
        const int lane = tid & 31;
        const int hl   = lane >> 4;        // K-pair select within WMMA step
        const int l    = lane & 15;        // M (A) / N (B) index within tile
        const int map  = wave >> 2;
        const int n0   = (wave & 3) * 16;
        // lane-resident base pointers: loop index becomes a pure immediate offset
        const v2f* ap = (const v2f*)Apk + hl * 16 + l;
        const v2f* bp = (const v2f*)(map ? Bpk2 : Bpk1) + hl * 64 + n0 + l;
        v8f acc = {};
#pragma unroll
        for (int k = 0; k < 32; ++k) {
            v2f a = ap[k * 32];            // ds_load_b64, offset = k*256 B
            v2f b = bp[k * 128];           // ds_load_b64, offset = k*1024 B
            acc = __builtin_amdgcn_wmma_f32_16x16x4_f32(
                false, a, false, b, (short)0, acc, false, false);
        }
#pragma unroll
        for (int g = 0; g < 8; ++g)
            Mm[map * 16 * NQ_PAD + (g + 8 * hl) * NQ_PAD + n0 + l] = acc[g];
    }
    __syncthreads();

    // ---- outputs: corr (E,512) | ac (E,2048) | valid (E) ----
    float* corr  = out;
    float* acout = out + (size_t)E * 512;
    float* vout  = out + (size_t)E * 512 + (size_t)E * 2048;

    for (int idx = tid; idx < 512; idx += 256) {
        int mp  = idx >> 8;
        int rem = idx & 255;
        int p   = rem >> 4;          // p = py*4 + px
        int rr  = rem & 15;          // r = ry*4 + rx
        int q   = ((p >> 2) + (rr >> 2)) * SEARCH + ((p & 3) + (rr & 3));
        corr[(size_t)e * 512 + idx] = Mm[mp * 16 * NQ_PAD + p * NQ_PAD + q] * vflg;
    }
    {
        const float* acg = patches_c + (size_t)ei * (C_CH * 16);
        for (int m = tid; m < C_CH * 16; m += 256)
            acout[(size_t)e * (C_CH * 16) + m] = acg[m];
    }
    if (tid == 0) vout[e] = vflg;
}

// ---------------- launch ----------------
extern "C" void kernel_launch(void* const* d_in, const int* in_sizes, int n_in,
                              void* d_out, int out_size, void* d_ws, size_t ws_size,
                              hipStream_t stream) {
    const float* poses       = (const float*)d_in[0];
    const float* patch_state = (const float*)d_in[1];
    const float* fmap        = (const float*)d_in[2];
    const float* patches_f   = (const float*)d_in[3];
    const float* patches_c   = (const float*)d_in[4];
    const int*   edge_i      = (const int*)d_in[5];
    const int*   edge_j      = (const int*)d_in[6];
    const int    E           = in_sizes[5];

    float* fmap2 = (float*)d_ws;   // 7*128*64*64 floats = 14.7 MB
    const int total2 = N_FRAMES * C_CH * H2 * W2;
    pool_kernel<<<(total2 + 255) / 256, 256, 0, stream>>>(fmap, fmap2, total2);

    const size_t smem_bytes = (size_t)SMEM_FLOATS * sizeof(float);
    corr_kernel<<<E, 256, smem_bytes, stream>>>(poses, patch_state, fmap,
                                                patches_f, patches_c, edge_i, edge_j,
                                                fmap2, (float*)d_out, E);
}